// CrossAttention_18528488915248
// MI455X (gfx1250) — compile-verified
//
#include <hip/hip_runtime.h>

#define NPAIRS 2048
#define DIMX 64
#define MMAX 128
#define NGROUPS 4096
#define PST 68      // stride for p/t/h tiles (128 x 64, padded)
#define SST 132     // stride for sim (128 x 128, padded)

typedef __attribute__((ext_vector_type(2))) float v2f;
typedef __attribute__((ext_vector_type(4))) float v4f;
typedef __attribute__((ext_vector_type(8))) float v8f;

__device__ __forceinline__ v8f wmma4(v2f a, v2f b, v8f c) {
  // D = A(16x4 f32) * B(4x16 f32) + C(16x16 f32)
  return __builtin_amdgcn_wmma_f32_16x16x4_f32(false, a, false, b, (short)0, c, false, false);
}

// ---------------- prefix-sum of group sizes (4096 entries) ----------------
__global__ void scan_sizes_kernel(const int* __restrict__ sizes, int* __restrict__ offs) {
  __shared__ int part[256];
  const int t = threadIdx.x;
  const int CH = NGROUPS / 256; // 16
  int s = 0;
  for (int i = 0; i < CH; ++i) s += sizes[t * CH + i];
  part[t] = s;
  __syncthreads();
  if (t == 0) {
    int acc = 0;
    for (int i = 0; i < 256; ++i) { int v = part[i]; part[i] = acc; acc += v; }
  }
  __syncthreads();
  int acc = part[t];
  for (int i = 0; i < CH; ++i) { offs[t * CH + i] = acc; acc += sizes[t * CH + i]; }
}

// ---------------- fused cross-attention: one block per pair ----------------
__global__ void __launch_bounds__(512, 1)
cross_attn_kernel(const float* __restrict__ data,
                  const float* __restrict__ W1, const float* __restrict__ b1,
                  const float* __restrict__ W2, const float* __restrict__ b2,
                  const int* __restrict__ sizes, const int* __restrict__ offs,
                  float* __restrict__ out_res, float* __restrict__ out_s1,
                  float* __restrict__ out_s2) {
  extern __shared__ float smem[];
  float* pA    = smem;                 // 128*68
  float* pB    = pA + MMAX * PST;      // 128*68
  float* tA    = pB + MMAX * PST;      // 128*68
  float* tB    = tA + MMAX * PST;      // 128*68
  float* simb  = tB + MMAX * PST;      // 128*132 (doubles as MLP hidden scratch)
  float* rowmx = simb + MMAX * SST;    // 128
  float* rowrc = rowmx + MMAX;         // 128 (reciprocal of row sum)
  float* colmx = rowrc + MMAX;         // 128
  float* colrc = colmx + MMAX;         // 128

  const int b   = blockIdx.x;
  const int tid = threadIdx.x;
  const int w   = tid >> 5;   // wave 0..15
  const int l   = tid & 31;
  const int lm  = l & 15;     // row/col within 16-tile
  const int hi  = l >> 4;     // half-wave -> K offset +2, row offset +8 for C/D

  const int n1 = sizes[2 * b], n2 = sizes[2 * b + 1];
  const int off1 = offs[2 * b], off2 = offs[2 * b + 1];

  // ---- Phase 1: gather p1, p2 into LDS, zero-padded ----
  for (int it = 0; it < 4; ++it) {
    int g = tid + it * 512;                 // 2048 float4 groups per side
    int row = g >> 4, c = (g & 15) * 4;
    v4f z = {0.f, 0.f, 0.f, 0.f};
    v4f v = z;
    if (row < n1) v = *(const v4f*)(data + ((size_t)(off1 + row)) * DIMX + c);
    *(v4f*)(pA + row * PST + c) = v;
    v = z;
    if (row < n2) v = *(const v4f*)(data + ((size_t)(off2 + row)) * DIMX + c);
    *(v4f*)(pB + row * PST + c) = v;
  }
  __syncthreads();

  // ---- Phase 2: two-layer MLP on both sides (WMMA f32, K=64 -> 16 steps) ----
  // 128x64 output = 8x4 tiles; wave w -> ntile = w&3, mtiles = (w>>2), (w>>2)+4
  {
    const int nt  = w & 3;
    const int mt0 = w >> 2;
    const int ncol = nt * 16 + lm;
    v2f w1f[16], w2f[16];
    for (int ks = 0; ks < 16; ++ks) {
      int k = ks * 4 + hi * 2;
      w1f[ks].x = W1[k * DIMX + ncol];
      w1f[ks].y = W1[(k + 1) * DIMX + ncol];
      w2f[ks].x = W2[k * DIMX + ncol];
      w2f[ks].y = W2[(k + 1) * DIMX + ncol];
    }
    const float bias1 = b1[ncol], bias2 = b2[ncol];
    float* H = simb;  // hidden scratch, stride PST

    for (int side = 0; side < 2; ++side) {
      const float* P = side ? pB : pA;
      float* T = side ? tB : tA;
      const int n = side ? n2 : n1;
      // layer 1: H = relu(P @ W1 + b1)
      for (int mi = 0; mi < 2; ++mi) {
        int mt = mt0 + mi * 4;
        int m = mt * 16 + lm;
        v8f acc = {0.f,0.f,0.f,0.f,0.f,0.f,0.f,0.f};
        for (int ks = 0; ks < 16; ++ks) {
          int k = ks * 4 + hi * 2;
          v2f a; a.x = P[m * PST + k]; a.y = P[m * PST + k + 1];
          acc = wmma4(a, w1f[ks], acc);
        }
        for (int v = 0; v < 8; ++v) {
          int row = mt * 16 + v + hi * 8;
          H[row * PST + ncol] = fmaxf(acc[v] + bias1, 0.f);
        }
      }
      __syncthreads();
      // layer 2: T = mask(H @ W2 + b2)
      for (int mi = 0; mi < 2; ++mi) {
        int mt = mt0 + mi * 4;
        int m = mt * 16 + lm;
        v8f acc = {0.f,0.f,0.f,0.f,0.f,0.f,0.f,0.f};
        for (int ks = 0; ks < 16; ++ks) {
          int k = ks * 4 + hi * 2;
          v2f a; a.x = H[m * PST + k]; a.y = H[m * PST + k + 1];
          acc = wmma4(a, w2f[ks], acc);
        }
        for (int v = 0; v < 8; ++v) {
          int row = mt * 16 + v + hi * 8;
          T[row * PST + ncol] = (row < n) ? (acc[v] + bias2) : 0.f;
        }
      }
      __syncthreads();
    }
  }

  // ---- Phase 3: sim = tA @ tB^T (128x128 = 8x8 tiles; 4 tiles/wave) + XOR mask ----
  {
    const int nt  = w & 7;
    const int mt0 = w >> 3;
    const int scol = nt * 16 + lm;
    const bool vn = scol < n2;
    for (int mi = 0; mi < 4; ++mi) {
      int mt = mt0 + mi * 2;
      int m = mt * 16 + lm;
      v8f acc = {0.f,0.f,0.f,0.f,0.f,0.f,0.f,0.f};
      for (int ks = 0; ks < 16; ++ks) {
        int k = ks * 4 + hi * 2;
        v2f a;  a.x  = tA[m * PST + k];    a.y  = tA[m * PST + k + 1];
        v2f bb; bb.x = tB[scol * PST + k]; bb.y = tB[scol * PST + k + 1]; // t2^T
        acc = wmma4(a, bb, acc);
      }
      for (int v = 0; v < 8; ++v) {
        int row = mt * 16 + v + hi * 8;
        bool vm = row < n1;
        simb[row * SST + scol] = (vm != vn) ? -__builtin_inff() : acc[v];
      }
    }
  }
  __syncthreads();

  // ---- Phase 4: softmax stats (rows: threads 0-127, cols: threads 128-255) ----
  if (tid < 256) {
    const int idx = tid & 127;
    if (tid < 128) {
      float mx = -__builtin_inff();
      for (int j = 0; j < MMAX; ++j) mx = fmaxf(mx, simb[idx * SST + j]);
      float s = 0.f;
      for (int j = 0; j < MMAX; ++j) s += __expf(simb[idx * SST + j] - mx);
      rowmx[idx] = mx; rowrc[idx] = 1.f / s;
    } else {
      float mx = -__builtin_inff();
      for (int i = 0; i < MMAX; ++i) mx = fmaxf(mx, simb[i * SST + idx]);
      float s = 0.f;
      for (int i = 0; i < MMAX; ++i) s += __expf(simb[i * SST + idx] - mx);
      colmx[idx] = mx; colrc[idx] = 1.f / s;
    }
  }
  __syncthreads();

  // ---- Phase 5: stream s1, s2 to global (nontemporal; read-only on LDS) ----
  {
    float* s1o = out_s1 + (size_t)b * MMAX * MMAX;
    float* s2o = out_s2 + (size_t)b * MMAX * MMAX;
    for (int g = tid; g < MMAX * 32; g += 512) {
      int row = g >> 5, c = (g & 31) * 4;
      float rm = rowmx[row], rr = rowrc[row];
      v4f o1, o2;
      for (int j = 0; j < 4; ++j) {
        float sv = simb[row * SST + c + j];
        o1[j] = __expf(sv - rm) * rr;
        o2[j] = __expf(sv - colmx[c + j]) * colrc[c + j];
      }
      __builtin_nontemporal_store(o1, (v4f*)(s1o + row * MMAX + c));
      __builtin_nontemporal_store(o2, (v4f*)(s2o + row * MMAX + c));
    }
  }

  // ---- Phase 6: attention GEMMs with on-the-fly softmax A operands (K=128) ----
  if (w < 8) {
    // query_new = s1 @ p2 ; waves 0-7, tiles: nt=w&3, mt = (w>>2)+{0,2,4,6}
    const int nt  = w & 3;
    const int mt0 = w >> 2;
    const int dcol = nt * 16 + lm;
    for (int mi = 0; mi < 4; ++mi) {
      int mt = mt0 + mi * 2;
      int m = mt * 16 + lm;
      float rm = rowmx[m], rr = rowrc[m];
      v8f acc = {0.f,0.f,0.f,0.f,0.f,0.f,0.f,0.f};
      for (int ks = 0; ks < 32; ++ks) {
        int k = ks * 4 + hi * 2;
        v2f a;
        a.x = __expf(simb[m * SST + k] - rm) * rr;
        a.y = __expf(simb[m * SST + k + 1] - rm) * rr;
        v2f bb; bb.x = pB[k * PST + dcol]; bb.y = pB[(k + 1) * PST + dcol];
        acc = wmma4(a, bb, acc);
      }
      for (int v = 0; v < 8; ++v) {
        int row = mt * 16 + v + hi * 8;
        if (row < n1) out_res[((size_t)(off1 + row)) * DIMX + dcol] = acc[v];
      }
    }
  } else {
    // corpus_new = s2^T @ p1 ; waves 8-15
    const int w2  = w - 8;
    const int nt  = w2 & 3;
    const int rt0 = w2 >> 2;
    const int dcol = nt * 16 + lm;
    for (int mi = 0; mi < 4; ++mi) {
      int rt = rt0 + mi * 2;
      int r = rt * 16 + lm;
      float cm = colmx[r], cr = colrc[r];
      v8f acc = {0.f,0.f,0.f,0.f,0.f,0.f,0.f,0.f};
      for (int ks = 0; ks < 32; ++ks) {
        int k = ks * 4 + hi * 2;
        v2f a;
        a.x = __expf(simb[k * SST + r] - cm) * cr;
        a.y = __expf(simb[(k + 1) * SST + r] - cm) * cr;
        v2f bb; bb.x = pA[k * PST + dcol]; bb.y = pA[(k + 1) * PST + dcol];
        acc = wmma4(a, bb, acc);
      }
      for (int v = 0; v < 8; ++v) {
        int row = rt * 16 + v + hi * 8;
        if (row < n2) out_res[((size_t)(off2 + row)) * DIMX + dcol] = acc[v];
      }
    }
  }
}

extern "C" void kernel_launch(void* const* d_in, const int* in_sizes, int n_in,
                              void* d_out, int out_size, void* d_ws, size_t ws_size,
                              hipStream_t stream) {
  const float* data = (const float*)d_in[0];
  const float* W1   = (const float*)d_in[1];
  const float* b1   = (const float*)d_in[2];
  const float* W2   = (const float*)d_in[3];
  const float* b2   = (const float*)d_in[4];
  const int*   sizes = (const int*)d_in[5];

  int* offs = (int*)d_ws;  // 4096 ints

  const int total = in_sizes[0] / DIMX;
  float* out_res = (float*)d_out;
  float* out_s1  = out_res + (size_t)total * DIMX;
  float* out_s2  = out_s1 + (size_t)NPAIRS * MMAX * MMAX;

  const size_t smem_bytes =
      (size_t)(4 * MMAX * PST + MMAX * SST + 4 * MMAX) * sizeof(float);  // ~204 KB
  (void)hipFuncSetAttribute((const void*)cross_attn_kernel,
                            hipFuncAttributeMaxDynamicSharedMemorySize,
                            (int)smem_bytes);

  scan_sizes_kernel<<<1, 256, 0, stream>>>(sizes, offs);
  cross_attn_kernel<<<NPAIRS, 512, smem_bytes, stream>>>(
      data, W1, b1, W2, b2, sizes, offs, out_res, out_s1, out_s2);
}